// GCN_17506286699046
// MI455X (gfx1250) — compile-verified
//
#include <hip/hip_runtime.h>
#include <hip/hip_bf16.h>

#define IN_SIZE 512
#define HIDDEN  16
#define OUT_SIZE 64

typedef __attribute__((ext_vector_type(2))) float v2f;
typedef __attribute__((ext_vector_type(4))) float v4f;
typedef __attribute__((ext_vector_type(8))) float v8f;

// ---------------- degree accumulation ----------------
__global__ void degree_kernel(const int* __restrict__ src, const int* __restrict__ dst,
                              float* __restrict__ degO, float* __restrict__ degI, int nE) {
    int e = blockIdx.x * blockDim.x + threadIdx.x;
    if (e >= nE) return;
    unsafeAtomicAdd(&degO[src[e]], 1.0f);
    unsafeAtomicAdd(&degI[dst[e]], 1.0f);
}

// in-place deg -> deg^-0.5 with clip(deg,1)
__global__ void norm_kernel(float* __restrict__ deg, int n) {
    int i = blockIdx.x * blockDim.x + threadIdx.x;
    if (i >= n) return;
    float d = deg[i];
    d = d < 1.0f ? 1.0f : d;
    deg[i] = 1.0f / sqrtf(d);
}

// ---------------- GEMM1: h1[n,16] = norm_src[n] * (x[n,:] @ W1) ----------------
// one wave per 16-row tile; fp32 WMMA 16x16x4 with permuted K-slots:
// macro-step i covers K=8i..8i+7 via one b128 A load + one b128 B load -> 2 WMMAs.
#define W1T_STRIDE 516  // padded to avoid LDS bank conflicts (516 % 64 = 4)
__global__ __launch_bounds__(256) void gemm1_kernel(
    const float* __restrict__ x, const float* __restrict__ W1,
    const float* __restrict__ norm_src, float* __restrict__ h1,
    int nNodes, int nTiles) {
    __shared__ float w1t[HIDDEN * W1T_STRIDE];  // W1 transposed: w1t[n][k], ~33 KB
    for (int i = threadIdx.x; i < IN_SIZE * HIDDEN; i += blockDim.x) {
        int k = i >> 4, n = i & 15;             // coalesced read of W1[k][n]
        w1t[n * W1T_STRIDE + k] = W1[i];
    }
    __syncthreads();

    int wave = threadIdx.x >> 5;
    int lane = threadIdx.x & 31;
    int tile = blockIdx.x * 8 + wave;
    if (tile >= nTiles) return;

    int half = lane >> 4;
    int lr   = lane & 15;
    int rowL = tile * 16 + lr;                  // logical A row
    int rowA = rowL < nNodes ? rowL : nNodes - 1;  // clamped: no divergence in loop
    const float* xrow = x + (long long)rowA * IN_SIZE;
    const float* brow = w1t + lr * W1T_STRIDE;

    v8f acc = {};
#pragma unroll 4
    for (int i = 0; i < IN_SIZE / 8; ++i) {
        int kb = 8 * i + 4 * half;              // half0: K=8i.., half1: K=8i+4..
        v4f af = __builtin_nontemporal_load((const v4f*)(xrow + kb));
        v4f bf = *(const v4f*)(brow + kb);
        v2f a1, b1, a2, b2;
        a1.x = af.x; a1.y = af.y; b1.x = bf.x; b1.y = bf.y;
        a2.x = af.z; a2.y = af.w; b2.x = bf.z; b2.y = bf.w;
        acc = __builtin_amdgcn_wmma_f32_16x16x4_f32(false, a1, false, b1, (short)0, acc, false, false);
        acc = __builtin_amdgcn_wmma_f32_16x16x4_f32(false, a2, false, b2, (short)0, acc, false, false);
    }
    // C layout: VGPR j -> row tile*16 + half*8 + j, col lr; scale by norm_src at store.
    // Wave-uniform full-tile fast path: no per-lane exec masking, vector norm loads.
    int rowC = tile * 16 + half * 8;
    if (tile * 16 + 16 <= nNodes) {
        v4f ns0 = *(const v4f*)(norm_src + rowC);
        v4f ns1 = *(const v4f*)(norm_src + rowC + 4);
        float* hp = h1 + (long long)rowC * HIDDEN + lr;
        hp[0 * HIDDEN] = acc[0] * ns0.x;
        hp[1 * HIDDEN] = acc[1] * ns0.y;
        hp[2 * HIDDEN] = acc[2] * ns0.z;
        hp[3 * HIDDEN] = acc[3] * ns0.w;
        hp[4 * HIDDEN] = acc[4] * ns1.x;
        hp[5 * HIDDEN] = acc[5] * ns1.y;
        hp[6 * HIDDEN] = acc[6] * ns1.z;
        hp[7 * HIDDEN] = acc[7] * ns1.w;
    } else {
#pragma unroll
        for (int j = 0; j < 8; ++j) {
            int r = rowC + j;
            if (r < nNodes) h1[(long long)r * HIDDEN + lr] = acc[j] * norm_src[r];
        }
    }
}

// ---------------- SpMM: agg[dst] += h[src], 16 feats, 4 threads/edge ----------------
__global__ void spmm_kernel(const int* __restrict__ src, const int* __restrict__ dst,
                            const float* __restrict__ h, float* __restrict__ agg, int nE) {
    long long tid = (long long)blockIdx.x * blockDim.x + threadIdx.x;
    long long e = tid >> 2;
    int p = (int)(tid & 3);
    if (e >= nE) return;
    int s = src[e], d = dst[e];
    v4f v = *(const v4f*)(h + (long long)s * HIDDEN + p * 4);
    float* ap = agg + (long long)d * HIDDEN + p * 4;
    unsafeAtomicAdd(ap + 0, v.x);
    unsafeAtomicAdd(ap + 1, v.y);
    unsafeAtomicAdd(ap + 2, v.z);
    unsafeAtomicAdd(ap + 3, v.w);
}

// ---------------- layer-1 tail fused with layer-2 head ----------------
// h2 = relu(agg1*norm_dst + b1) * norm_src   (one float4 per thread)
__global__ void post1_kernel(const float* __restrict__ agg1, const float* __restrict__ b1,
                             const float* __restrict__ norm_src, const float* __restrict__ norm_dst,
                             float* __restrict__ h2, int nNodes) {
    int i = blockIdx.x * blockDim.x + threadIdx.x;   // over nNodes*4 quads
    if (i >= nNodes * 4) return;
    int n = i >> 2, q = (i & 3) * 4;
    float nd = norm_dst[n], ns = norm_src[n];
    v4f v = *(const v4f*)(agg1 + (long long)n * HIDDEN + q);
    v4f bb = *(const v4f*)(b1 + q);
    v4f r;
    r.x = fmaxf(v.x * nd + bb.x, 0.0f) * ns;
    r.y = fmaxf(v.y * nd + bb.y, 0.0f) * ns;
    r.z = fmaxf(v.z * nd + bb.z, 0.0f) * ns;
    r.w = fmaxf(v.w * nd + bb.w, 0.0f) * ns;
    *(v4f*)(h2 + (long long)n * HIDDEN + q) = r;
}

// ---------------- GEMM2: out[n,64] = (agg2[n,:] @ W2) * norm_dst[n] + b2 ----------------
#define W2T_STRIDE 20   // 16 + pad
__global__ __launch_bounds__(256) void gemm2_kernel(
    const float* __restrict__ agg2, const float* __restrict__ W2,
    const float* __restrict__ b2, const float* __restrict__ norm_dst,
    float* __restrict__ out, int nNodes, int nTiles) {
    __shared__ float w2t[OUT_SIZE * W2T_STRIDE];  // W2 transposed: w2t[n][k], 5 KB
    __shared__ float b2s[OUT_SIZE];
    for (int i = threadIdx.x; i < HIDDEN * OUT_SIZE; i += blockDim.x) {
        int k = i >> 6, n = i & 63;               // coalesced read of W2[k][n]
        w2t[n * W2T_STRIDE + k] = W2[i];
    }
    if (threadIdx.x < OUT_SIZE) b2s[threadIdx.x] = b2[threadIdx.x];
    __syncthreads();

    int wave = threadIdx.x >> 5;
    int lane = threadIdx.x & 31;
    int tile = blockIdx.x * 8 + wave;
    if (tile >= nTiles) return;

    int half = lane >> 4;
    int lr   = lane & 15;
    int rowL = tile * 16 + lr;
    int rowA = rowL < nNodes ? rowL : nNodes - 1;
    const float* arow = agg2 + (long long)rowA * HIDDEN;

    v8f acc0 = {}, acc1 = {}, acc2 = {}, acc3 = {};
#pragma unroll
    for (int i = 0; i < HIDDEN / 8; ++i) {        // 2 macro-steps of 8 K
        int kb = 8 * i + 4 * half;
        v4f af = *(const v4f*)(arow + kb);
        v2f a1, a2;
        a1.x = af.x; a1.y = af.y; a2.x = af.z; a2.y = af.w;
#pragma unroll
        for (int t = 0; t < 4; ++t) {
            v4f bf = *(const v4f*)(w2t + (t * 16 + lr) * W2T_STRIDE + kb);
            v2f b1f, b2f;
            b1f.x = bf.x; b1f.y = bf.y; b2f.x = bf.z; b2f.y = bf.w;
            v8f* accp = t == 0 ? &acc0 : t == 1 ? &acc1 : t == 2 ? &acc2 : &acc3;
            *accp = __builtin_amdgcn_wmma_f32_16x16x4_f32(false, a1, false, b1f, (short)0, *accp, false, false);
            *accp = __builtin_amdgcn_wmma_f32_16x16x4_f32(false, a2, false, b2f, (short)0, *accp, false, false);
        }
    }

    int rowC = tile * 16 + half * 8;
    float blo = b2s[lr], b16 = b2s[16 + lr], b32v = b2s[32 + lr], b48 = b2s[48 + lr];
    if (tile * 16 + 16 <= nNodes) {
        v4f nd0 = *(const v4f*)(norm_dst + rowC);
        v4f nd1 = *(const v4f*)(norm_dst + rowC + 4);
        float ndv[8] = {nd0.x, nd0.y, nd0.z, nd0.w, nd1.x, nd1.y, nd1.z, nd1.w};
#pragma unroll
        for (int j = 0; j < 8; ++j) {
            float nd = ndv[j];
            float* orow = out + (long long)(rowC + j) * OUT_SIZE;
            orow[0  + lr] = acc0[j] * nd + blo;
            orow[16 + lr] = acc1[j] * nd + b16;
            orow[32 + lr] = acc2[j] * nd + b32v;
            orow[48 + lr] = acc3[j] * nd + b48;
        }
    } else {
#pragma unroll
        for (int j = 0; j < 8; ++j) {
            int r = rowC + j;
            if (r >= nNodes) continue;
            float nd = norm_dst[r];
            float* orow = out + (long long)r * OUT_SIZE;
            orow[0  + lr] = acc0[j] * nd + blo;
            orow[16 + lr] = acc1[j] * nd + b16;
            orow[32 + lr] = acc2[j] * nd + b32v;
            orow[48 + lr] = acc3[j] * nd + b48;
        }
    }
}

extern "C" void kernel_launch(void* const* d_in, const int* in_sizes, int n_in,
                              void* d_out, int out_size, void* d_ws, size_t ws_size,
                              hipStream_t stream) {
    const float* x   = (const float*)d_in[0];
    const float* W1  = (const float*)d_in[1];
    const float* b1  = (const float*)d_in[2];
    const float* W2  = (const float*)d_in[3];
    const float* b2  = (const float*)d_in[4];
    const int*   src = (const int*)d_in[5];
    const int*   dst = (const int*)d_in[6];
    const int nE = in_sizes[5];
    const int nN = in_sizes[0] / IN_SIZE;
    float* out = (float*)d_out;

    // workspace layout: [norm_src nN][norm_dst nN][bufA nN*16][bufB nN*16]
    float* norm_src = (float*)d_ws;
    float* norm_dst = norm_src + nN;
    float* bufA     = norm_dst + nN;                  // h1, then h2
    float* bufB     = bufA + (size_t)nN * HIDDEN;     // agg1, then agg2

    const int nTiles = (nN + 15) / 16;
    const int nBlkT  = (nTiles + 7) / 8;

    // degrees -> norms
    (void)hipMemsetAsync(norm_src, 0, sizeof(float) * 2 * (size_t)nN, stream);
    degree_kernel<<<(nE + 255) / 256, 256, 0, stream>>>(src, dst, norm_src, norm_dst, nE);
    norm_kernel<<<(2 * nN + 255) / 256, 256, 0, stream>>>(norm_src, 2 * nN);

    // layer 1: transform first (512 -> 16), then aggregate
    gemm1_kernel<<<nBlkT, 256, 0, stream>>>(x, W1, norm_src, bufA, nN, nTiles);
    (void)hipMemsetAsync(bufB, 0, sizeof(float) * (size_t)nN * HIDDEN, stream);
    {
        long long th = (long long)nE * 4;
        spmm_kernel<<<(unsigned)((th + 255) / 256), 256, 0, stream>>>(src, dst, bufA, bufB, nE);
    }
    post1_kernel<<<(nN * 4 + 255) / 256, 256, 0, stream>>>(bufB, b1, norm_src, norm_dst, bufA, nN);

    // layer 2: aggregate first, then transform (16 -> 64)
    (void)hipMemsetAsync(bufB, 0, sizeof(float) * (size_t)nN * HIDDEN, stream);
    {
        long long th = (long long)nE * 4;
        spmm_kernel<<<(unsigned)((th + 255) / 256), 256, 0, stream>>>(src, dst, bufA, bufB, nE);
    }
    gemm2_kernel<<<nBlkT, 256, 0, stream>>>(bufB, W2, b2, norm_dst, out, nN, nTiles);
}